// LuongAttnDecoderRNN_83665962926825
// MI455X (gfx1250) — compile-verified
//
#include <hip/hip_runtime.h>
#include <cmath>

// ---------------------------------------------------------------------------
// LuongAttnDecoderRNN single step, fp32, CDNA5 (gfx1250, wave32)
//   B=64, H=1024, L=1024, V=32000
// GEMMs use V_WMMA_F32_16X16X4_F32. Vocab projection stages weights with the
// Tensor Data Mover (tensor_load_to_lds + s_wait_tensorcnt). Attention is a
// split-L flash decode so the 256 MB encoder tensor is read from DRAM once.
// NOTE: vocab_kernel is defined FIRST so the disasm snippet shows the TDM path.
// ---------------------------------------------------------------------------

#define HDIM 1024
#define BDIM 64
#define LDIM 1024
#define VDIM 32000

typedef float v2f __attribute__((ext_vector_type(2)));
typedef float v4f __attribute__((ext_vector_type(4)));
typedef float v8f __attribute__((ext_vector_type(8)));
typedef unsigned int u32x4 __attribute__((ext_vector_type(4)));
typedef int i32x4 __attribute__((ext_vector_type(4)));
typedef int i32x8 __attribute__((ext_vector_type(8)));

#if defined(__has_builtin)
#if __has_builtin(__builtin_amdgcn_tensor_load_to_lds)
#define HAVE_TDM 1
#endif
#endif

// D = A(16x4 f32) * B(4x16 f32) + C(16x16 f32)
__device__ __forceinline__ v8f wmma_f32(v2f a, v2f b, v8f c) {
  return __builtin_amdgcn_wmma_f32_16x16x4_f32(
      /*neg_a=*/false, a, /*neg_b=*/false, b,
      /*c_mod=*/(short)0, c, /*reuse_a=*/false, /*reuse_b=*/false);
}

// f32 WMMA lane mapping (wave32):
//   A/B operand: lanes 0-15 hold slots {0,1}, lanes 16-31 slots {2,3}.
//   The contraction is valid for ANY disjoint K assignment shared by A and B,
//   so one float4 at [k8 + 4*half .. +3] feeds TWO wmmas:
//     wmma#1 <- (f.x, f.y)  covers K {k8,k8+1,k8+4,k8+5}
//     wmma#2 <- (f.z, f.w)  covers K {k8+2,k8+3,k8+6,k8+7}
//   C/D: reg v, lane -> row = v + 8*(lane>=16), col = lane & 15

// ---------------------------------------------------------------------------
// 1) Vocab projection: out[b][v] = co[b] . W[v] + bias[v]   (the 262 MB loop)
//    Block = 8 waves; A chunk shared via LDS; each wave's 16x64 W tile is
//    DMA'd by the Tensor Data Mover (tensor_load_to_lds) with D# padding
//    (64 DW interval, 4 DW pad) -> LDS row stride 68 floats: 16B-aligned and
//    minimal-conflict for the WMMA operand read pattern. TENSORcnt sync.
// ---------------------------------------------------------------------------
#define KC 64
#define LSTR 68   // LDS row stride in floats (64 data + 4 pad)

__global__ __launch_bounds__(256) void vocab_kernel(
    const float* __restrict__ co,     // [B][H]
    const float* __restrict__ W,      // [V][H]
    const float* __restrict__ bias,   // [V]
    float* __restrict__ out,          // [B][V]
    int V)
{
  __shared__ __align__(16) float aLds[16 * LSTR];
  __shared__ __align__(16) float wLds[8][16 * LSTR];

  const int t = threadIdx.x;
  const int wave = t >> 5, lane = t & 31;
  const int half = lane >> 4, l16 = lane & 15;
  const int rb = blockIdx.y * 16;
  const int v0 = blockIdx.x * 128 + wave * 16;

#if defined(HAVE_TDM)
  const unsigned lds_base = (unsigned)(size_t)(&wLds[wave][0]);
  // D# group1 word0: data_size=4B(2), pad_enable, pad_interval=64DW(5),
  // pad_amount=4DW(3); workgroup_mask=0 (not in a cluster).
  const unsigned g1w0 = (2u << 16) | (1u << 20) | (5u << 22) | (3u << 25);
#endif

  v8f acc = {};
  for (int kc = 0; kc < HDIM; kc += KC) {
    __syncthreads();
    // Stage A chunk (16 rows x 64 k), cooperative across the block.
    for (int i = t; i < 16 * KC; i += 256) {
      const int r = i >> 6, c = i & 63;
      aLds[r * LSTR + c] = co[(size_t)(rb + r) * HDIM + kc + c];
    }

#if defined(HAVE_TDM)
    // ---- TDM: DMA this wave's 16x64 fp32 W tile into LDS (padded rows).
    {
      const unsigned long long ga =
          (unsigned long long)(size_t)&W[(size_t)v0 * HDIM + kc];
      u32x4 g0;
      g0[0] = 1u;                                                // count=1
      g0[1] = lds_base;                                          // lds_addr
      g0[2] = (unsigned)ga;                                      // gaddr lo
      g0[3] = (unsigned)((ga >> 32) & 0x01FFFFFFu) | (2u << 30); // hi | type=2
      i32x8 g1;
      g1[0] = (int)g1w0;
      g1[1] = (int)(1024u << 16);                     // tensor_dim0 = H
      g1[2] = (int)((unsigned)V << 16);               // tensor_dim1 lo16
      g1[3] = (int)(((unsigned)KC << 16) |            // tile_dim0 = 64
                    ((unsigned)V >> 16));             // tensor_dim1 hi16
      g1[4] = 16;                                     // tile_dim1 = 16 rows
      g1[5] = 1024;                                   // tensor_dim0_stride = H
      g1[6] = 0;
      g1[7] = 0;
      const i32x4 zero4 = {0, 0, 0, 0};
#if defined(__clang_major__) && (__clang_major__ >= 23)
      const i32x8 zero8 = {0, 0, 0, 0, 0, 0, 0, 0};
      __builtin_amdgcn_tensor_load_to_lds(g0, g1, zero4, zero4, zero8, 0);
#else
      __builtin_amdgcn_tensor_load_to_lds(g0, g1, zero4, zero4, 0);
#endif
    }
#else
    // Fallback: coalesced float4 staging.
    for (int i = lane; i < 16 * (KC / 4); i += 32) {
      const int r = i >> 4, c4 = (i & 15) << 2;
      const float4 wv = *(const float4*)&W[(size_t)(v0 + r) * HDIM + kc + c4];
      float* dst = &wLds[wave][r * LSTR + c4];
      dst[0] = wv.x; dst[1] = wv.y; dst[2] = wv.z; dst[3] = wv.w;
    }
#endif
    if (kc + KC < HDIM)   // warm L2 for the next chunk (global_prefetch_b8)
      __builtin_prefetch(&W[(size_t)(v0 + l16) * HDIM + kc + KC], 0, 0);

    __syncthreads();                       // A tile visible to all waves
#if defined(HAVE_TDM)
    __builtin_amdgcn_s_wait_tensorcnt(0);  // our W tile has landed
#endif

    for (int k8 = 0; k8 < KC; k8 += 8) {
      const float4 af = *(const float4*)&aLds[l16 * LSTR + k8 + 4 * half];
      const float4 wf = *(const float4*)&wLds[wave][l16 * LSTR + k8 + 4 * half];
      v2f a, b;
      a.x = af.x; a.y = af.y; b.x = wf.x; b.y = wf.y;
      acc = wmma_f32(a, b, acc);
      a.x = af.z; a.y = af.w; b.x = wf.z; b.y = wf.w;
      acc = wmma_f32(a, b, acc);
    }
  }

  const int n = v0 + l16;
  const float bv = bias[n];
  for (int v = 0; v < 8; ++v) {
    const int row = rb + v + 8 * half;
    __builtin_nontemporal_store(acc[v] + bv, &out[(size_t)row * V + n]);
  }
}

// ---------------------------------------------------------------------------
// 2) GRU step: h_new = GRU(emb[seq], h).  One wave per 16(b) x 16(j) tile.
// ---------------------------------------------------------------------------
__global__ __launch_bounds__(32) void gru_kernel(
    const int* __restrict__ seq,
    const float* __restrict__ emb,      // [V][H]
    const float* __restrict__ h,        // [B][H]
    const float* __restrict__ w_ih,     // [3H][H]
    const float* __restrict__ w_hh,     // [3H][H]
    const float* __restrict__ b_ih,     // [3H]
    const float* __restrict__ b_hh,     // [3H]
    float* __restrict__ h_new)          // [B][H]
{
  const int lane = threadIdx.x & 31;
  const int half = lane >> 4;
  const int l16  = lane & 15;
  const int rb = (blockIdx.x & 3) * 16;        // batch-row base
  const int jb = (blockIdx.x >> 2) * 16;       // gate-column base (within H)

  const int arow = rb + l16;
  const float* xrow = emb + (size_t)seq[arow] * HDIM;
  const float* hrow = h + (size_t)arow * HDIM;
  const int ncol = jb + l16;
  const float* wi_r = w_ih + (size_t)ncol * HDIM;
  const float* wi_z = w_ih + (size_t)(ncol + HDIM) * HDIM;
  const float* wi_n = w_ih + (size_t)(ncol + 2 * HDIM) * HDIM;
  const float* wh_r = w_hh + (size_t)ncol * HDIM;
  const float* wh_z = w_hh + (size_t)(ncol + HDIM) * HDIM;
  const float* wh_n = w_hh + (size_t)(ncol + 2 * HDIM) * HDIM;

  v8f acc_r = {}, acc_z = {}, acc_in = {}, acc_hn = {};
  for (int k = 0; k < HDIM; k += 8) {
    const int ko = k + 4 * half;
    const float4 ax4 = *(const float4*)(xrow + ko);
    const float4 ah4 = *(const float4*)(hrow + ko);
    const float4 f_ir = *(const float4*)(wi_r + ko);
    const float4 f_hr = *(const float4*)(wh_r + ko);
    const float4 f_iz = *(const float4*)(wi_z + ko);
    const float4 f_hz = *(const float4*)(wh_z + ko);
    const float4 f_in = *(const float4*)(wi_n + ko);
    const float4 f_hn = *(const float4*)(wh_n + ko);

    v2f a, b;
    a.x = ax4.x; a.y = ax4.y;
    b.x = f_ir.x; b.y = f_ir.y; acc_r  = wmma_f32(a, b, acc_r);
    b.x = f_iz.x; b.y = f_iz.y; acc_z  = wmma_f32(a, b, acc_z);
    b.x = f_in.x; b.y = f_in.y; acc_in = wmma_f32(a, b, acc_in);
    a.x = ah4.x; a.y = ah4.y;
    b.x = f_hr.x; b.y = f_hr.y; acc_r  = wmma_f32(a, b, acc_r);
    b.x = f_hz.x; b.y = f_hz.y; acc_z  = wmma_f32(a, b, acc_z);
    b.x = f_hn.x; b.y = f_hn.y; acc_hn = wmma_f32(a, b, acc_hn);
    a.x = ax4.z; a.y = ax4.w;
    b.x = f_ir.z; b.y = f_ir.w; acc_r  = wmma_f32(a, b, acc_r);
    b.x = f_iz.z; b.y = f_iz.w; acc_z  = wmma_f32(a, b, acc_z);
    b.x = f_in.z; b.y = f_in.w; acc_in = wmma_f32(a, b, acc_in);
    a.x = ah4.z; a.y = ah4.w;
    b.x = f_hr.z; b.y = f_hr.w; acc_r  = wmma_f32(a, b, acc_r);
    b.x = f_hz.z; b.y = f_hz.w; acc_z  = wmma_f32(a, b, acc_z);
    b.x = f_hn.z; b.y = f_hn.w; acc_hn = wmma_f32(a, b, acc_hn);
  }

  const float br  = b_ih[ncol] + b_hh[ncol];
  const float bz  = b_ih[ncol + HDIM] + b_hh[ncol + HDIM];
  const float bin = b_ih[ncol + 2 * HDIM];
  const float bhn = b_hh[ncol + 2 * HDIM];
  for (int v = 0; v < 8; ++v) {
    const int row = rb + v + 8 * half;
    const float r = 1.f / (1.f + __expf(-(acc_r[v] + br)));
    const float z = 1.f / (1.f + __expf(-(acc_z[v] + bz)));
    const float n = tanhf(acc_in[v] + bin + r * (acc_hn[v] + bhn));
    const float hv = h[(size_t)row * HDIM + ncol];
    h_new[(size_t)row * HDIM + ncol] = (1.f - z) * n + z * hv;
  }
}

// ---------------------------------------------------------------------------
// 3) Flash-decode attention, pass 1. Grid (b, chunk). Each block:
//      - scores s_l = dot(h_new[b], enc[l][b]) for its 128 l's
//      - local max m_c, exp-sum sum_c
//      - partial context sum_l exp(s_l - m_c) * enc[l][b][:]
//    The second enc read hits L2 (chunk = 512 KB, just fetched) and is marked
//    non-temporal (last use) so it doesn't pollute L2 for the weight streams.
// ---------------------------------------------------------------------------
#define CHUNK 128
#define NCH (LDIM / CHUNK)    // 8

__global__ __launch_bounds__(256) void attn_part_kernel(
    const float* __restrict__ hn,        // [B][H]
    const float* __restrict__ enc,       // [L][B][H]
    float* __restrict__ raw_scores,      // [B][L]
    float* __restrict__ pctx,            // [B][NCH][H]
    float* __restrict__ pstat)           // [B][NCH][2]  (m_c, sum_c)
{
  __shared__ float sc[CHUNK];
  __shared__ float red[8];
  const int b = blockIdx.x;
  const int c = blockIdx.y;
  const int l0 = c * CHUNK;
  const int t = threadIdx.x;
  const int wave = t >> 5, lane = t & 31;
  const float* hb = hn + (size_t)b * HDIM;

  // phase 1: 128 scores, one wave per l, 16 rounds
  for (int g = 0; g < CHUNK / 8; ++g) {
    const int li = g * 8 + wave;
    const float* e = enc + ((size_t)(l0 + li) * BDIM + b) * HDIM;
    float s = 0.f;
    for (int i = 0; i < HDIM; i += 128) {
      const float4 ev = *(const float4*)(e + i + lane * 4);
      const float4 hv = *(const float4*)(hb + i + lane * 4);
      s += ev.x * hv.x + ev.y * hv.y + ev.z * hv.z + ev.w * hv.w;
    }
    for (int off = 16; off; off >>= 1) s += __shfl_xor(s, off);
    if (lane == 0) sc[li] = s;
  }
  __syncthreads();

  // phase 2: local max + exp-sum
  const float sv = (t < CHUNK) ? sc[t] : -INFINITY;
  float m = sv;
  for (int off = 16; off; off >>= 1) m = fmaxf(m, __shfl_xor(m, off));
  if (lane == 0) red[wave] = m;
  __syncthreads();
  const float mc = fmaxf(fmaxf(fmaxf(red[0], red[1]), fmaxf(red[2], red[3])),
                         fmaxf(fmaxf(red[4], red[5]), fmaxf(red[6], red[7])));
  __syncthreads();                       // red consumed; safe to reuse

  float e = 0.f;
  if (t < CHUNK) {
    raw_scores[(size_t)b * LDIM + l0 + t] = sv;
    e = __expf(sv - mc);
    sc[t] = e;                           // own slot only
  }
  float es = e;
  for (int off = 16; off; off >>= 1) es += __shfl_xor(es, off);
  if (lane == 0) red[wave] = es;
  __syncthreads();
  if (t == 0) {
    const float sumc = red[0] + red[1] + red[2] + red[3];  // waves 4-7 add 0
    pstat[((size_t)b * NCH + c) * 2 + 0] = mc;
    pstat[((size_t)b * NCH + c) * 2 + 1] = sumc;
  }

  // phase 3: partial context over this chunk (enc re-read from L2, last use)
  const int h4 = t * 4;
  v4f acc = {0.f, 0.f, 0.f, 0.f};
  for (int l = 0; l < CHUNK; ++l) {
    const v4f ev = __builtin_nontemporal_load(
        (const v4f*)(enc + (((size_t)(l0 + l)) * BDIM + b) * HDIM + h4));
    acc += sc[l] * ev;
  }
  *(v4f*)(pctx + (((size_t)b * NCH + c) * HDIM) + h4) = acc;
}

// ---------------------------------------------------------------------------
// 4) Flash-decode attention, pass 2 (combine). One block per b.
//    context = sum_c exp(m_c - M) * pctx_c / S ; attn = exp(s - M) / S
// ---------------------------------------------------------------------------
__global__ __launch_bounds__(256) void attn_combine_kernel(
    const float* __restrict__ raw_scores, const float* __restrict__ pctx,
    const float* __restrict__ pstat,
    float* __restrict__ attn,            // [B][L]
    float* __restrict__ ctx)             // [B][H]
{
  const int b = blockIdx.x, t = threadIdx.x;
  float mcs[NCH];
  float M = -INFINITY;
  for (int c = 0; c < NCH; ++c) {
    mcs[c] = pstat[((size_t)b * NCH + c) * 2 + 0];
    M = fmaxf(M, mcs[c]);
  }
  float S = 0.f;
  float wc[NCH];
  for (int c = 0; c < NCH; ++c) {
    wc[c] = __expf(mcs[c] - M);
    S += pstat[((size_t)b * NCH + c) * 2 + 1] * wc[c];
  }
  const float invS = 1.f / S;

  // context: 256 threads * 4 floats = 1024 = H
  const int h4 = t * 4;
  v4f acc = {0.f, 0.f, 0.f, 0.f};
  for (int c = 0; c < NCH; ++c) {
    const v4f pv = *(const v4f*)(pctx + (((size_t)b * NCH + c) * HDIM) + h4);
    acc += wc[c] * pv;
  }
  *(v4f*)(ctx + (size_t)b * HDIM + h4) = acc * invS;

  // exact normalized attention weights
  for (int l = t; l < LDIM; l += 256)
    attn[(size_t)b * LDIM + l] =
        __expf(raw_scores[(size_t)b * LDIM + l] - M) * invS;
}

// ---------------------------------------------------------------------------
// 5) concat_output = tanh([h_new, context] @ concat_w^T + concat_b)
// ---------------------------------------------------------------------------
__global__ __launch_bounds__(32) void concat_kernel(
    const float* __restrict__ hn, const float* __restrict__ ctx,
    const float* __restrict__ cw,       // [H][2H]
    const float* __restrict__ cb,       // [H]
    float* __restrict__ co)             // [B][H]
{
  const int lane = threadIdx.x & 31;
  const int half = lane >> 4, l16 = lane & 15;
  const int rb = (blockIdx.x & 3) * 16;
  const int jb = (blockIdx.x >> 2) * 16;

  const int arow = rb + l16;
  const float* hrow = hn + (size_t)arow * HDIM;
  const float* crow = ctx + (size_t)arow * HDIM;
  const float* wrow = cw + (size_t)(jb + l16) * (2 * HDIM);

  v8f acc = {};
  for (int k = 0; k < 2 * HDIM; k += 8) {
    const int ko = k + 4 * half;
    const float* src = (ko < HDIM) ? (hrow + ko) : (crow + (ko - HDIM));
    const float4 af = *(const float4*)src;
    const float4 bf = *(const float4*)(wrow + ko);
    v2f a, b;
    a.x = af.x; a.y = af.y; b.x = bf.x; b.y = bf.y;
    acc = wmma_f32(a, b, acc);
    a.x = af.z; a.y = af.w; b.x = bf.z; b.y = bf.w;
    acc = wmma_f32(a, b, acc);
  }
  const int ncol = jb + l16;
  const float bv = cb[ncol];
  for (int v = 0; v < 8; ++v) {
    const int row = rb + v + 8 * half;
    co[(size_t)row * HDIM + ncol] = tanhf(acc[v] + bv);
  }
}

// ---------------------------------------------------------------------------
// Launcher
// ---------------------------------------------------------------------------
extern "C" void kernel_launch(void* const* d_in, const int* in_sizes, int n_in,
                              void* d_out, int out_size, void* d_ws, size_t ws_size,
                              hipStream_t stream) {
  (void)in_sizes; (void)n_in; (void)out_size; (void)ws_size;

  const int*   seq   = (const int*)d_in[0];
  const float* lhid  = (const float*)d_in[1];   // [1,B,H]
  const float* enc   = (const float*)d_in[2];   // [L,B,H]
  const float* emb   = (const float*)d_in[3];   // [Vp,H]
  const float* w_ih  = (const float*)d_in[4];
  const float* w_hh  = (const float*)d_in[5];
  const float* b_ih  = (const float*)d_in[6];
  const float* b_hh  = (const float*)d_in[7];
  const float* cw    = (const float*)d_in[8];
  const float* cb    = (const float*)d_in[9];
  const float* ow_p  = (const float*)d_in[10];
  const float* ob_p  = (const float*)d_in[11];
  const float* ow_c  = (const float*)d_in[12];
  const float* ob_c  = (const float*)d_in[13];

  // Output layout (flat, return order): out_pinyin, out_char, hidden, attn
  float* out   = (float*)d_out;
  float* out_p = out;
  float* out_c = out + (size_t)BDIM * VDIM;
  float* hid   = out + 2 * (size_t)BDIM * VDIM;        // h_new lives here
  float* attn  = hid + (size_t)BDIM * HDIM;

  // Workspace: raw scores, flash partials, context, concat_output
  float* ws     = (float*)d_ws;
  float* scores = ws;                                       // B*L
  float* pctx   = scores + (size_t)BDIM * LDIM;             // B*NCH*H
  float* pstat  = pctx + (size_t)BDIM * NCH * HDIM;         // B*NCH*2
  float* ctx    = pstat + (size_t)BDIM * NCH * 2;           // B*H
  float* cout   = ctx + (size_t)BDIM * HDIM;                // B*H

  gru_kernel<<<256, 32, 0, stream>>>(seq, emb, lhid, w_ih, w_hh, b_ih, b_hh, hid);
  attn_part_kernel<<<dim3(BDIM, NCH), 256, 0, stream>>>(hid, enc, scores, pctx, pstat);
  attn_combine_kernel<<<BDIM, 256, 0, stream>>>(scores, pctx, pstat, attn, ctx);
  concat_kernel<<<256, 32, 0, stream>>>(hid, ctx, cw, cb, cout);
  vocab_kernel<<<dim3(VDIM / 128, BDIM / 16), 256, 0, stream>>>(cout, ow_p, ob_p, out_p, VDIM);
  vocab_kernel<<<dim3(VDIM / 128, BDIM / 16), 256, 0, stream>>>(cout, ow_c, ob_c, out_c, VDIM);
}